// DoubleAttentionPairBlock_63943473103364
// MI455X (gfx1250) — compile-verified
//
#include <hip/hip_runtime.h>
#include <hip/hip_bf16.h>

// ---------------------------------------------------------------------------
// DoubleAttentionPairBlock for MI455X (gfx1250), f32 end-to-end.
// Memory-bound problem (151MB edge tensor, 23.3 TB/s HBM, 192MB L2):
// two passes over P + one output write ~= 450MB ~= 19us floor.
// All GEMMs use V_WMMA_F32_16X16X4_F32 (full f32 precision; FLOPs are
// negligible vs bandwidth, so the f32 WMMA path is the right precision).
//
// f32 16x16x4 WMMA fragment layout assumptions (from CDNA5 ISA 7.12.2):
//   A (16xK): lane 0-15 -> M=lane,  V0=K0,V1=K1 ; lane 16-31 -> M=lane-16, V0=K2,V1=K3
//   B (Kx16): lane 0-15 -> N=lane,  V0=K0,V1=K1 ; lane 16-31 -> N=lane-16, V0=K2,V1=K3
//   C/D     : reg j, lane L -> M = j + 8*(L>=16), N = L&15
// ---------------------------------------------------------------------------

typedef float v2f __attribute__((ext_vector_type(2)));
typedef float v8f __attribute__((ext_vector_type(8)));

#define BB 2
#define NN 384
#define CC 128
#define DD 32
#define MM (NN * NN)        // 147456 rows per batch
#define TILES (MM / 16)     // 9216 row-tiles per batch
#define GBX 96              // blocks in x per batch
#define WPB 8               // waves per block (256 threads, wave32)
#define WAVES (GBX * WPB)   // 768 waves per batch
#define ITERS (TILES / WAVES) // 12, exact

__device__ __forceinline__ v8f wmma4(v2f a, v2f b, v8f c) {
  // D = A(16x4,f32) * B(4x16,f32) + C(16x16,f32)
  return __builtin_amdgcn_wmma_f32_16x16x4_f32(
      /*neg_a=*/false, a, /*neg_b=*/false, b,
      /*c_mod=*/(short)0, c, /*reuse_a=*/false, /*reuse_b=*/false);
}

// Regs in C-layout hold X[m][x] with (m = j + 8*half, x = l15). Return, to
// every lane, X[row=m][x = this lane's l15]  (cross-half gather via bpermute).
template <int m>
__device__ __forceinline__ float bcast_row(const float E[8], int l15) {
  float x;
  if constexpr (m < 8) x = E[m];        // owner: lanes 0-15
  else                 x = E[m - 8];    // owner: lanes 16-31
  return __shfl(x, l15 + ((m < 8) ? 0 : 16), 32);
}

// Build an A- or B-fragment (K-group ks, K0=4*ks) of the *transposed* view of
// a C-layout tile: frag.V0 = X[K = 4ks + 2*half][this lane's l15], V1 = K+1.
template <int ks>
__device__ __forceinline__ v2f fragT(const float E[8], int l15, int half) {
  float lo0 = bcast_row<4 * ks + 0>(E, l15);
  float lo1 = bcast_row<4 * ks + 1>(E, l15);
  float hi0 = bcast_row<4 * ks + 2>(E, l15);
  float hi1 = bcast_row<4 * ks + 3>(E, l15);
  v2f r;
  r.x = half ? hi0 : lo0;
  r.y = half ? hi1 : lo1;
  return r;
}

// ---------------------------------------------------------------------------
__global__ void kz_kernel(float* ws) {
  int i = blockIdx.x * 256 + threadIdx.x;
  if (i < 2112) ws[i] = 0.0f;  // numer (2048) + den (64)
}

// ---------------------------------------------------------------------------
// Pass A: fP=P@Wg+bg, gP=P@Wv+bv, accumulate den[d]=sum_m exp(fP[m,d]) and
// numer[d][e]=sum_m exp(fP[m,d])*gP[m,e].  (exp without max-shift: fP~N(0,1),
// max over 147k samples ~5 -> exp(5)=148, sum ~2.4e5: safely inside f32.)
__global__ __launch_bounds__(256) void ka_kernel(
    const float* __restrict__ P, const float* __restrict__ Wg,
    const float* __restrict__ bg, const float* __restrict__ Wv,
    const float* __restrict__ bv, float* __restrict__ numer,
    float* __restrict__ den) {
  __shared__ __align__(16) float WgT[DD * CC];  // Wg^T [32][128], contiguous K
  __shared__ __align__(16) float WvT[DD * CC];
  __shared__ float lnum[DD * DD];
  __shared__ float lden[DD];
  const int tid = threadIdx.x;
  for (int i = tid; i < CC * DD; i += 256) {
    int c = i >> 5, d = i & 31;           // Wg row-major [128][32]
    WgT[d * CC + c] = Wg[i];
    WvT[d * CC + c] = Wv[i];
  }
  for (int i = tid; i < DD * DD; i += 256) lnum[i] = 0.0f;
  if (tid < DD) lden[tid] = 0.0f;
  __syncthreads();

  const int batch = blockIdx.y;
  const float* Pb = P + (size_t)batch * MM * CC;
  const int wave = tid >> 5, lane = tid & 31, l15 = lane & 15, half = lane >> 4;
  const int gw = blockIdx.x * WPB + wave;

  v8f accN[2][2] = {};   // numer tiles: [d-tile][e-tile]
  float denAcc = 0.0f;
  const float bg0 = bg[l15], bg1 = bg[l15 + 16];
  const float bv0 = bv[l15], bv1 = bv[l15 + 16];

  for (int it = 0; it < ITERS; ++it) {
    const int tIdx = gw + it * WAVES;
    const float* Prow = Pb + (size_t)(tIdx * 16 + l15) * CC;
    v8f aF0 = {}, aF1 = {}, aG0 = {}, aG1 = {};
#pragma unroll
    for (int s = 0; s < 32; ++s) {
      const int k0 = 4 * s + 2 * half;
      v2f a = *(const v2f*)(Prow + k0);                    // A frag of P tile
      v2f bgA = *(const v2f*)(&WgT[l15 * CC + k0]);        // B frags, cols 0-15
      v2f bgB = *(const v2f*)(&WgT[(l15 + 16) * CC + k0]); // cols 16-31
      v2f bvA = *(const v2f*)(&WvT[l15 * CC + k0]);
      v2f bvB = *(const v2f*)(&WvT[(l15 + 16) * CC + k0]);
      aF0 = wmma4(a, bgA, aF0);
      aF1 = wmma4(a, bgB, aF1);
      aG0 = wmma4(a, bvA, aG0);
      aG1 = wmma4(a, bvB, aG1);
    }
    // bias + exp; E/G in C-layout regs
    float E[2][8], G[2][8];
#pragma unroll
    for (int j = 0; j < 8; ++j) {
      E[0][j] = expf(aF0[j] + bg0);
      E[1][j] = expf(aF1[j] + bg1);
      G[0][j] = aG0[j] + bv0;
      G[1][j] = aG1[j] + bv1;
    }
    // den partial: this lane's d = lane (l15 + 16*half)
    float se0 = 0.0f, se1 = 0.0f;
#pragma unroll
    for (int j = 0; j < 8; ++j) { se0 += E[0][j]; se1 += E[1][j]; }
    se0 += __shfl_xor(se0, 16, 32);
    se1 += __shfl_xor(se1, 16, 32);
    denAcc += half ? se1 : se0;
    // numer += E^T (32d x 16m) @ G (16m x 32e): 4 K-groups, 4 output tiles
#define NUMER_STEP(ks)                               \
  {                                                  \
    v2f aT0 = fragT<ks>(E[0], l15, half);            \
    v2f aT1 = fragT<ks>(E[1], l15, half);            \
    v2f bT0 = fragT<ks>(G[0], l15, half);            \
    v2f bT1 = fragT<ks>(G[1], l15, half);            \
    accN[0][0] = wmma4(aT0, bT0, accN[0][0]);        \
    accN[0][1] = wmma4(aT0, bT1, accN[0][1]);        \
    accN[1][0] = wmma4(aT1, bT0, accN[1][0]);        \
    accN[1][1] = wmma4(aT1, bT1, accN[1][1]);        \
  }
    NUMER_STEP(0) NUMER_STEP(1) NUMER_STEP(2) NUMER_STEP(3)
#undef NUMER_STEP
  }
  // wave -> block (LDS atomics), block -> global (global atomics)
#pragma unroll
  for (int td = 0; td < 2; ++td)
#pragma unroll
    for (int te = 0; te < 2; ++te)
#pragma unroll
      for (int j = 0; j < 8; ++j) {
        int d = 16 * td + j + 8 * half;
        int e = 16 * te + l15;
        atomicAdd(&lnum[d * DD + e], accN[td][te][j]);
      }
  atomicAdd(&lden[lane], denAcc);
  __syncthreads();
  for (int i = tid; i < DD * DD; i += 256)
    atomicAdd(&numer[batch * DD * DD + i], lnum[i]);
  if (tid < DD) atomicAdd(&den[batch * DD + tid], lden[tid]);
}

// ---------------------------------------------------------------------------
// Finalize V (stored transposed: VT[e][d]), FiLM MLP, and Wo^T staging.
__global__ __launch_bounds__(256) void kb_kernel(
    const float* __restrict__ numer, const float* __restrict__ den,
    float* __restrict__ VT, const float* __restrict__ r3t,
    const float* __restrict__ Wt1, const float* __restrict__ bt1,
    const float* __restrict__ Wt2, const float* __restrict__ bt2,
    float* __restrict__ film, const float* __restrict__ Wo,
    float* __restrict__ WoT) {
  const int b = blockIdx.x, tid = threadIdx.x;
  for (int i = tid; i < DD * DD; i += 256) {
    int d = i >> 5, e = i & 31;
    VT[b * DD * DD + e * DD + d] = numer[b * DD * DD + i] / den[b * DD + d];
  }
  __shared__ float h[128];
  const float x = r3t[b];
  if (tid < 128) {
    float z = x * Wt1[tid] + bt1[tid];
    h[tid] = 0.5f * z * (1.0f + erff(z * 0.70710678118654752f));  // exact GELU
  }
  __syncthreads();
  float acc = bt2[tid];  // tid in [0,256): 0-127 -> scale, 128-255 -> shift
  for (int j = 0; j < 128; ++j) acc += h[j] * Wt2[j * 256 + tid];
  film[b * 256 + tid] = acc;
  if (b == 0)
    for (int i = tid; i < DD * CC; i += 256) {
      int d = i >> 7, c = i & 127;  // Wo row-major [32][128]
      WoT[c * DD + d] = Wo[i];
    }
}

// ---------------------------------------------------------------------------
// Pass C: scores=rowsoftmax(P@Wd+bd); O=scores@V; out=P+(O@Wo+bo)*(1+s)+t.
__global__ __launch_bounds__(256) void kc_kernel(
    const float* __restrict__ P, const float* __restrict__ Wd,
    const float* __restrict__ bd, const float* __restrict__ VT,
    const float* __restrict__ WoT, const float* __restrict__ bo,
    const float* __restrict__ film, float* __restrict__ out) {
  __shared__ __align__(16) float WdT[DD * CC];
  __shared__ __align__(16) float SB[WPB][16 * DD];  // per-wave transpose buf
  const int tid = threadIdx.x;
  for (int i = tid; i < CC * DD; i += 256) {
    int c = i >> 5, d = i & 31;
    WdT[d * CC + c] = Wd[i];
  }
  __syncthreads();
  const int batch = blockIdx.y;
  const float* Pb = P + (size_t)batch * MM * CC;
  float* Ob = out + (size_t)batch * MM * CC;
  const float* VTb = VT + batch * DD * DD;
  const float* scl = film + batch * 256;
  const float* shf = scl + 128;
  const int wave = tid >> 5, lane = tid & 31, l15 = lane & 15, half = lane >> 4;
  const int gw = blockIdx.x * WPB + wave;
  float* S = SB[wave];
  const float bd0 = bd[l15], bd1 = bd[l15 + 16];

  for (int it = 0; it < ITERS; ++it) {
    const int tIdx = gw + it * WAVES;
    const float* Prow = Pb + (size_t)(tIdx * 16 + l15) * CC;
    v8f aS0 = {}, aS1 = {};
#pragma unroll
    for (int s = 0; s < 32; ++s) {
      const int k0 = 4 * s + 2 * half;
      v2f a = *(const v2f*)(Prow + k0);
      v2f b0 = *(const v2f*)(&WdT[l15 * CC + k0]);
      v2f b1 = *(const v2f*)(&WdT[(l15 + 16) * CC + k0]);
      aS0 = wmma4(a, b0, aS0);
      aS1 = wmma4(a, b1, aS1);
    }
    // row softmax over 32 channels (lanes within 16-group + both tiles)
    float sc0[8], sc1[8];
#pragma unroll
    for (int j = 0; j < 8; ++j) {
      float v0 = aS0[j] + bd0, v1 = aS1[j] + bd1;
      float mx = fmaxf(v0, v1);
      mx = fmaxf(mx, __shfl_xor(mx, 1, 32));
      mx = fmaxf(mx, __shfl_xor(mx, 2, 32));
      mx = fmaxf(mx, __shfl_xor(mx, 4, 32));
      mx = fmaxf(mx, __shfl_xor(mx, 8, 32));
      float e0 = expf(v0 - mx), e1 = expf(v1 - mx);
      float sm = e0 + e1;
      sm += __shfl_xor(sm, 1, 32);
      sm += __shfl_xor(sm, 2, 32);
      sm += __shfl_xor(sm, 4, 32);
      sm += __shfl_xor(sm, 8, 32);
      float inv = 1.0f / sm;
      sc0[j] = e0 * inv;
      sc1[j] = e1 * inv;
    }
    // C-layout -> row-major LDS [16m][32d] so it can be re-read as A frags
#pragma unroll
    for (int j = 0; j < 8; ++j) {
      S[(j + 8 * half) * DD + l15] = sc0[j];
      S[(j + 8 * half) * DD + l15 + 16] = sc1[j];
    }
    asm volatile("" ::: "memory");  // keep DS store->load order (in-order DS)
    v8f o0 = {}, o1 = {};
#pragma unroll
    for (int ks = 0; ks < 8; ++ks) {
      const int k0 = 4 * ks + 2 * half;
      v2f a = *(const v2f*)(&S[l15 * DD + k0]);
      v2f b0 = *(const v2f*)(&VTb[l15 * DD + k0]);
      v2f b1 = *(const v2f*)(&VTb[(l15 + 16) * DD + k0]);
      o0 = wmma4(a, b0, o0);
      o1 = wmma4(a, b1, o1);
    }
    asm volatile("" ::: "memory");
#pragma unroll
    for (int j = 0; j < 8; ++j) {  // O tile -> LDS [16m][32e]
      S[(j + 8 * half) * DD + l15] = o0[j];
      S[(j + 8 * half) * DD + l15 + 16] = o1[j];
    }
    asm volatile("" ::: "memory");
    v8f accC[8] = {};
#pragma unroll
    for (int ks = 0; ks < 8; ++ks) {
      const int k0 = 4 * ks + 2 * half;
      v2f a = *(const v2f*)(&S[l15 * DD + k0]);
#pragma unroll
      for (int tc = 0; tc < 8; ++tc) {
        v2f bb = *(const v2f*)(&WoT[(16 * tc + l15) * DD + k0]);
        accC[tc] = wmma4(a, bb, accC[tc]);
      }
    }
    // fused bias + FiLM + residual epilogue
#pragma unroll
    for (int tc = 0; tc < 8; ++tc) {
      const int c = 16 * tc + l15;
      const float s1 = 1.0f + scl[c], sh = shf[c], bov = bo[c];
#pragma unroll
      for (int j = 0; j < 8; ++j) {
        const size_t m = (size_t)(tIdx * 16 + j + 8 * half);
        Ob[m * CC + c] = Pb[m * CC + c] + (accC[tc][j] + bov) * s1 + sh;
      }
    }
  }
}

// ---------------------------------------------------------------------------
extern "C" void kernel_launch(void* const* d_in, const int* in_sizes, int n_in,
                              void* d_out, int out_size, void* d_ws,
                              size_t ws_size, hipStream_t stream) {
  (void)in_sizes; (void)n_in; (void)out_size; (void)ws_size;
  const float* edge = (const float*)d_in[0];
  const float* r3t = (const float*)d_in[1];
  const float* Wg = (const float*)d_in[2];
  const float* bg = (const float*)d_in[3];
  const float* Wv = (const float*)d_in[4];
  const float* bv = (const float*)d_in[5];
  const float* Wd = (const float*)d_in[6];
  const float* bd = (const float*)d_in[7];
  const float* Wo = (const float*)d_in[8];
  const float* bo = (const float*)d_in[9];
  const float* Wt1 = (const float*)d_in[10];
  const float* bt1 = (const float*)d_in[11];
  const float* Wt2 = (const float*)d_in[12];
  const float* bt2 = (const float*)d_in[13];
  float* ws = (float*)d_ws;
  float* numer = ws;          // 2048 floats
  float* den = ws + 2048;     // 64
  float* VT = ws + 2112;      // 2048
  float* film = ws + 4160;    // 512
  float* WoT = ws + 4672;     // 4096
  float* outp = (float*)d_out;

  kz_kernel<<<dim3(9), dim3(256), 0, stream>>>(ws);
  ka_kernel<<<dim3(GBX, BB), dim3(256), 0, stream>>>(edge, Wg, bg, Wv, bv,
                                                     numer, den);
  kb_kernel<<<dim3(BB), dim3(256), 0, stream>>>(numer, den, VT, r3t, Wt1, bt1,
                                                Wt2, bt2, film, Wo, WoT);
  kc_kernel<<<dim3(GBX, BB), dim3(256), 0, stream>>>(edge, Wd, bd, VT, WoT, bo,
                                                     film, outp);
}